// Net_25288767439580
// MI455X (gfx1250) — compile-verified
//
#include <hip/hip_runtime.h>

// ---------------------------------------------------------------------------
// SINDy autoencoder forward, collapsed linear chains + WMMA f32 16x16x4.
//   z   = x  @ Jenc^T + benc          Jenc = We3@We2@We1   [3,128]
//   dz  = dx @ Jenc^T
//   dzb = theta(z) @ EW^T + Eb
//   xb  = z   @ Jdec^T + bdec         Jdec = Wd3@Wd2@Wd1   [128,3]
//   dxb = dzb @ Jdec^T
// Outputs concatenated: z | dz | dzb | xb | dxb  (float32)
// ---------------------------------------------------------------------------

typedef float v2f __attribute__((ext_vector_type(2)));
typedef float v8f __attribute__((ext_vector_type(8)));

// Workspace layout (floats)
#define WS_BENCP 0      // packed Benc: [64 kpair][16 n][2]  = 2048
#define WS_BENC  2048   // effective encoder bias, padded    = 4
#define WS_BDEC  2052   // effective decoder bias            = 128
#define WS_BDECP 2180   // packed Bdec: [2 kpair][128 n][2]  = 512
// total 2692 floats

// ---------------------------------------------------------------------------
// Stage 1: fold weight chains, pack WMMA B-fragments.
// Benc[k][n] = (n<3) ? Jenc[n][k] : 0        (B is 128x16, K-major pairs)
// BencP[(k>>1)*32 + n*2 + (k&1)] = Benc[k][n]
// BdecP[(h*128 + d)*2 + c] = (2h+c < 3) ? Jdec[d][2h+c] : 0
// ---------------------------------------------------------------------------
__global__ __launch_bounds__(128) void sindy_prep(
    const float* __restrict__ We1, const float* __restrict__ be1,
    const float* __restrict__ We2, const float* __restrict__ be2,
    const float* __restrict__ We3, const float* __restrict__ be3,
    const float* __restrict__ Wd1, const float* __restrict__ bd1,
    const float* __restrict__ Wd2, const float* __restrict__ bd2,
    const float* __restrict__ Wd3, const float* __restrict__ bd3,
    float* __restrict__ ws)
{
    __shared__ float T1[192];   // We3@We2            [3][64]
    __shared__ float T2[192];   // Wd2@Wd1            [64][3]
    __shared__ float v32[32];   // We2@be1 + be2      [32]
    __shared__ float w64[64];   // Wd2@bd1 + bd2      [64]
    const int t = threadIdx.x;

    for (int idx = t; idx < 192; idx += 128) {
        int i = idx >> 6, j = idx & 63;
        float s = 0.f;
        for (int k = 0; k < 32; ++k) s += We3[i*32+k] * We2[k*64+j];
        T1[i*64+j] = s;
    }
    if (t < 32) {
        float s = be2[t];
        for (int j = 0; j < 64; ++j) s += We2[t*64+j] * be1[j];
        v32[t] = s;
    }
    for (int idx = t; idx < 192; idx += 128) {
        int j = idx / 3, i = idx - 3*j;
        float s = 0.f;
        for (int k = 0; k < 32; ++k) s += Wd2[j*32+k] * Wd1[k*3+i];
        T2[j*3+i] = s;
    }
    if (t < 64) {
        float s = bd2[t];
        for (int k = 0; k < 32; ++k) s += Wd2[t*32+k] * bd1[k];
        w64[t] = s;
    }
    __syncthreads();

    if (t < 3) {
        float s = be3[t];
        for (int k = 0; k < 32; ++k) s += We3[t*32+k] * v32[k];
        ws[WS_BENC + t] = s;
    }
    if (t == 3) ws[WS_BENC + 3] = 0.f;

    {   // encoder: thread t == column d of Jenc
        const int d = t;
        float j0 = 0.f, j1 = 0.f, j2 = 0.f;
        for (int j = 0; j < 64; ++j) {
            float w = We1[j*128 + d];
            j0 += T1[j]      * w;
            j1 += T1[64 + j] * w;
            j2 += T1[128+ j] * w;
        }
        const int base = WS_BENCP + (d >> 1)*32 + (d & 1);
        ws[base + 0] = j0;
        ws[base + 2] = j1;
        ws[base + 4] = j2;
        for (int n = 3; n < 16; ++n) ws[base + 2*n] = 0.f;
    }
    {   // decoder: thread t == row d of Jdec
        const int d = t;
        float j0 = 0.f, j1 = 0.f, j2 = 0.f, bb = bd3[d];
        for (int j = 0; j < 64; ++j) {
            float w = Wd3[d*64 + j];
            j0 += w * T2[j*3 + 0];
            j1 += w * T2[j*3 + 1];
            j2 += w * T2[j*3 + 2];
            bb += w * w64[j];
        }
        ws[WS_BDEC + d] = bb;
        ws[WS_BDECP + d*2 + 0]       = j0;  // kpair 0: K=0,1
        ws[WS_BDECP + d*2 + 1]       = j1;
        ws[WS_BDECP + 256 + d*2 + 0] = j2;  // kpair 1: K=2,3(pad)
        ws[WS_BDECP + 256 + d*2 + 1] = 0.f;
    }
}

// ---------------------------------------------------------------------------
// Stage 2: streaming kernel. Block = 256 threads = 8 waves; 128 rows/block.
// LDS: x-tile + dx-tile [128][132] (pad -> conflict-free A-fragment b64 reads)
//      + per-wave z / dzb scratch [8][16][4].
// ---------------------------------------------------------------------------
#define LDS_STRIDE 132
#define SMEM_FLOATS (2*128*LDS_STRIDE + 1024)

__global__ __launch_bounds__(256) void sindy_main(
    const float* __restrict__ x, const float* __restrict__ dx,
    const float* __restrict__ EW, const float* __restrict__ Eb,
    const float* __restrict__ ws,
    float* __restrict__ z_out, float* __restrict__ dz_out,
    float* __restrict__ dzb_out, float* __restrict__ xb_out,
    float* __restrict__ dxb_out)
{
    extern __shared__ float smem[];
    float* xs  = smem;                          // [128][132]
    float* dxs = smem + 128*LDS_STRIDE;         // [128][132]
    float* zsm = smem + 2*128*LDS_STRIDE;       // [8][16][4]
    float* zbs = zsm + 512;                     // [8][16][4]

    const int tid  = threadIdx.x;
    const int row0 = blockIdx.x * 128;

    if (tid == 0) {   // warm packed weight tables (global_prefetch_b8)
        __builtin_prefetch(ws, 0, 0);
        __builtin_prefetch(ws + 1024, 0, 0);
        __builtin_prefetch(ws + 2048, 0, 0);
    }

    // -------- coalesced tile stage: 4096 float4 per tile, 16 per thread
    const float4* __restrict__ x4  = (const float4*)(x  + (size_t)row0 * 128);
    const float4* __restrict__ dx4 = (const float4*)(dx + (size_t)row0 * 128);
    for (int i = tid; i < 4096; i += 256) {
        int r = i >> 5, c = i & 31;
        float4 a = x4[i];
        float4 b = dx4[i];
        *(float4*)&xs [r*LDS_STRIDE + c*4] = a;
        *(float4*)&dxs[r*LDS_STRIDE + c*4] = b;
    }
    __syncthreads();

    const int lane  = tid & 31;
    const int wave  = tid >> 5;
    const int m16   = lane & 15;   // WMMA: A row / B-C-D column
    const int hi    = lane >> 4;   // lane-half selects K(-pair) / row+8
    const int rwave = row0 + wave * 16;

    // -------- encoder: 16x16x4 f32 WMMA, K = 128, cols 0..2 live
    v8f cz  = {0.f,0.f,0.f,0.f,0.f,0.f,0.f,0.f};
    v8f cdz = {0.f,0.f,0.f,0.f,0.f,0.f,0.f,0.f};
    const float* __restrict__ bencP = ws + WS_BENCP;
    const float* xrow  = xs  + (wave*16 + m16) * LDS_STRIDE;
    const float* dxrow = dxs + (wave*16 + m16) * LDS_STRIDE;
    #pragma unroll 4
    for (int k0 = 0; k0 < 128; k0 += 4) {
        const int kk = k0 + 2*hi;
        v2f a  = *(const v2f*)(xrow  + kk);                       // A: x[m][kk..kk+1]
        v2f ad = *(const v2f*)(dxrow + kk);
        v2f b  = *(const v2f*)(bencP + (kk >> 1)*32 + m16*2);     // B: Benc[kk..kk+1][n]
        cz  = __builtin_amdgcn_wmma_f32_16x16x4_f32(false, a,  false, b, (short)0, cz,  false, false);
        cdz = __builtin_amdgcn_wmma_f32_16x16x4_f32(false, ad, false, b, (short)0, cdz, false, false);
    }

    const float bn = (m16 < 3) ? ws[WS_BENC + m16] : 0.f;
    #pragma unroll
    for (int v = 0; v < 8; ++v) {
        const int m   = v + 8*hi;                 // C VGPR v -> rows v / v+8
        const int row = rwave + m;
        const float zv = cz[v] + bn;
        if (m16 < 3) {
            z_out [(size_t)row*3 + m16] = zv;
            dz_out[(size_t)row*3 + m16] = cdz[v];
        }
        if (m16 < 4) zsm[wave*64 + m*4 + m16] = zv;  // col 3 is exact 0 (padded B)
    }
    __syncthreads();

    // -------- SINDy library + dzb (tiny): lanes 0..15 own one row each
    if (lane < 16) {
        const float z0 = zsm[wave*64 + lane*4 + 0];
        const float z1 = zsm[wave*64 + lane*4 + 1];
        const float z2 = zsm[wave*64 + lane*4 + 2];
        const float q00=z0*z0, q01=z0*z1, q02=z0*z2, q11=z1*z1, q12=z1*z2, q22=z2*z2;
        const float th[22] = {1.f,1.f,1.f, z0,z1,z2,
                              q00,q01,q02,q11,q12,q22,
                              q00*z0,q00*z1,q00*z2,q01*z1,q01*z2,q02*z2,
                              q11*z1,q11*z2,q12*z2,q22*z2};
        const int row = rwave + lane;
        #pragma unroll
        for (int l = 0; l < 3; ++l) {
            float acc = Eb[l];
            #pragma unroll
            for (int t = 0; t < 22; ++t) acc += EW[l*22 + t] * th[t];
            dzb_out[(size_t)row*3 + l] = acc;
            zbs[wave*64 + lane*4 + l] = acc;
        }
        zbs[wave*64 + lane*4 + 3] = 0.f;
    }
    __syncthreads();

    // -------- decoder: one 16x16x4 WMMA per 16-col chunk (K=4, col3 padded 0)
    const int kz = 2*hi;
    v2f az  = *(const v2f*)(zsm + wave*64 + m16*4 + kz);
    v2f azb = *(const v2f*)(zbs + wave*64 + m16*4 + kz);
    const float* __restrict__ bdecP = ws + WS_BDECP;
    const v8f czero = {0.f,0.f,0.f,0.f,0.f,0.f,0.f,0.f};
    #pragma unroll
    for (int nc = 0; nc < 8; ++nc) {
        const int col = nc*16 + m16;
        v2f b = *(const v2f*)(bdecP + (hi*128 + col)*2);
        v8f cx  = __builtin_amdgcn_wmma_f32_16x16x4_f32(false, az,  false, b, (short)0, czero, false, false);
        v8f cdx = __builtin_amdgcn_wmma_f32_16x16x4_f32(false, azb, false, b, (short)0, czero, false, false);
        const float bb = ws[WS_BDEC + col];
        #pragma unroll
        for (int v = 0; v < 8; ++v) {
            const size_t row = (size_t)(rwave + v + 8*hi);
            xb_out [row*128 + col] = cx[v] + bb;
            dxb_out[row*128 + col] = cdx[v];
        }
    }
}

// ---------------------------------------------------------------------------
extern "C" void kernel_launch(void* const* d_in, const int* in_sizes, int n_in,
                              void* d_out, int out_size, void* d_ws, size_t ws_size,
                              hipStream_t stream)
{
    (void)n_in; (void)out_size; (void)ws_size;
    const float* x   = (const float*)d_in[0];
    const float* dx  = (const float*)d_in[1];
    const float* We1 = (const float*)d_in[2];
    const float* be1 = (const float*)d_in[3];
    const float* We2 = (const float*)d_in[4];
    const float* be2 = (const float*)d_in[5];
    const float* We3 = (const float*)d_in[6];
    const float* be3 = (const float*)d_in[7];
    const float* Wd1 = (const float*)d_in[8];
    const float* bd1 = (const float*)d_in[9];
    const float* Wd2 = (const float*)d_in[10];
    const float* bd2 = (const float*)d_in[11];
    const float* Wd3 = (const float*)d_in[12];
    const float* bd3 = (const float*)d_in[13];
    const float* EW  = (const float*)d_in[14];
    const float* Eb  = (const float*)d_in[15];

    float* ws = (float*)d_ws;
    const int N = in_sizes[0] / 128;          // 262144

    float* out     = (float*)d_out;
    float* z_out   = out;
    float* dz_out  = out + (size_t)N * 3;
    float* dzb_out = out + (size_t)N * 6;
    float* xb_out  = out + (size_t)N * 9;
    float* dxb_out = out + (size_t)N * 9 + (size_t)N * 128;

    sindy_prep<<<1, 128, 0, stream>>>(We1, be1, We2, be2, We3, be3,
                                      Wd1, bd1, Wd2, bd2, Wd3, bd3, ws);

    const int smemBytes = SMEM_FLOATS * (int)sizeof(float);   // 139264 B (<= 320KB WGP LDS)
    hipFuncSetAttribute((const void*)sindy_main,
                        hipFuncAttributeMaxDynamicSharedMemorySize, smemBytes);
    sindy_main<<<N / 128, 256, smemBytes, stream>>>(x, dx, EW, Eb, ws,
                                                    z_out, dz_out, dzb_out,
                                                    xb_out, dxb_out);
}